// MWT_CZ3d_49709951484272
// MI455X (gfx1250) — compile-verified
//
#include <hip/hip_runtime.h>
#include <math.h>

typedef float v2f __attribute__((ext_vector_type(2)));
typedef float v8f __attribute__((ext_vector_type(8)));

#define TWO_PI 6.283185307179586f

__device__ __forceinline__ int mode_k(int m, int n) {
    // mode set along a spatial axis of length n: {0..l1-1} U {n-l1..n-1}
    if (n <= 10) return m;
    return (m < 5) ? m : (n - 10 + m);
}

// ---------------------------------------------------------------------------
// Wavelet decomposition: xa gather (4 quadrants x 9ch = K36) then two GEMMs
// [M,36]x[36,9] -> d, s.  fp32 WMMA 16x16x4, one 16-row tile per wave.
// B matrices staged zero-padded in LDS -> branchless fragment loads.
// ---------------------------------------------------------------------------
__global__ void k_wavelet(const float* __restrict__ x,
                          const float* __restrict__ ecd,
                          const float* __restrict__ ecs,
                          float* __restrict__ dout,
                          float* __restrict__ sout,
                          int n, int tiles) {
    __shared__ float sD[36 * 16];
    __shared__ float sS[36 * 16];
    for (int idx = threadIdx.x; idx < 36 * 16; idx += blockDim.x) {
        int k = idx >> 4, c = idx & 15;
        float vd = 0.f, vs = 0.f;
        if (c < 9) { vd = ecd[k * 9 + c]; vs = ecs[k * 9 + c]; }
        sD[idx] = vd; sS[idx] = vs;
    }
    __syncthreads();

    int wid = blockIdx.x * (blockDim.x >> 5) + (threadIdx.x >> 5);
    if (wid < tiles) {
        int lane = threadIdx.x & 31;
        bool hi  = lane >= 16;
        int col  = lane & 15;
        int m0   = wid << 4;

        // A-fragment row owned by this lane
        int mrow = m0 + col;
        int t = mrow & 63;
        int r = mrow >> 6;
        int j = r % n; r /= n;
        int i = r % n;
        int b = r / n;
        int n2 = n << 1;

        v8f accd = {};
        v8f accs = {};
#pragma unroll
        for (int s = 0; s < 9; ++s) {       // K = 36 -> 9 steps of 4
            int kb = s * 4 + (hi ? 2 : 0);
            v2f a, bd, bs;
#pragma unroll
            for (int u = 0; u < 2; ++u) {
                int k = kb + u;
                int q = k / 9, c = k - q * 9;   // quadrant, channel
                int xi = 2 * i + (q >> 1);
                int yj = 2 * j + (q & 1);
                float av = x[(((size_t)(b * n2 + xi) * n2 + yj) * 64 + t) * 9 + c];
                float wd = sD[k * 16 + col];
                float ws = sS[k * 16 + col];
                if (u == 0) { a.x = av; bd.x = wd; bs.x = ws; }
                else        { a.y = av; bd.y = wd; bs.y = ws; }
            }
            accd = __builtin_amdgcn_wmma_f32_16x16x4_f32(false, a, false, bd, (short)0, accd, false, false);
            accs = __builtin_amdgcn_wmma_f32_16x16x4_f32(false, a, false, bs, (short)0, accs, false, false);
        }
        if (col < 9) {
#pragma unroll
            for (int q2 = 0; q2 < 8; ++q2) {
                int m = m0 + q2 + (hi ? 8 : 0);
                dout[(size_t)m * 9 + col] = accd[q2];
                sout[(size_t)m * 9 + col] = accs[q2];
            }
        }
    }
}

// ---------------------------------------------------------------------------
// Reconstruction: A = [x+Us | Ud] (K=18 padded to 20), four GEMMs with rc_*,
// fused even/odd interleave scatter to the 2n x 2n grid.  fp32 WMMA.
// rc_* staged zero-padded in LDS; A loads branchless (clamped idx + select).
// ---------------------------------------------------------------------------
__global__ void k_evenodd(const float* __restrict__ xp,
                          const float* __restrict__ us,
                          const float* __restrict__ ud,
                          const float* __restrict__ ree,
                          const float* __restrict__ reo,
                          const float* __restrict__ roe,
                          const float* __restrict__ roo,
                          float* __restrict__ out,
                          int n, int tiles) {
    __shared__ float sRC[4 * 20 * 16];      // [which][k][col], zero padded
    for (int idx = threadIdx.x; idx < 4 * 20 * 16; idx += blockDim.x) {
        int w = idx / 320, rem = idx - w * 320;
        int k = rem >> 4, c = rem & 15;
        const float* src = (w == 0) ? ree : (w == 1) ? reo : (w == 2) ? roe : roo;
        sRC[idx] = (k < 18 && c < 9) ? src[k * 9 + c] : 0.f;
    }
    __syncthreads();

    int wid = blockIdx.x * (blockDim.x >> 5) + (threadIdx.x >> 5);
    if (wid < tiles) {
        int lane = threadIdx.x & 31;
        bool hi  = lane >= 16;
        int col  = lane & 15;
        int m0   = wid << 4;
        int mrow = m0 + col;
        size_t rowb = (size_t)mrow * 9;

        v8f aee = {}, aeo = {}, aoe = {}, aoo = {};
#pragma unroll
        for (int s = 0; s < 5; ++s) {       // K = 18 -> pad 20 -> 5 steps
            int kb = s * 4 + (hi ? 2 : 0);
            v2f a, bee, beo, boe, boo;
#pragma unroll
            for (int u = 0; u < 2; ++u) {
                int k = kb + u;
                bool isX = k < 9;
                bool live = k < 18;
                int kc = live ? (isX ? k : k - 9) : 0;   // clamped, always valid
                const float* bp = isX ? xp : ud;          // pointer select (cndmask)
                float v1 = bp[rowb + kc];                 // unconditional load
                float v2 = us[rowb + kc];                 // unconditional load
                float va = v1 + (isX ? v2 : 0.f);
                va = live ? va : 0.f;
                float wee = sRC[0 * 320 + k * 16 + col];
                float weo = sRC[1 * 320 + k * 16 + col];
                float woe = sRC[2 * 320 + k * 16 + col];
                float woo = sRC[3 * 320 + k * 16 + col];
                if (u == 0) { a.x = va; bee.x = wee; beo.x = weo; boe.x = woe; boo.x = woo; }
                else        { a.y = va; bee.y = wee; beo.y = weo; boe.y = woe; boo.y = woo; }
            }
            aee = __builtin_amdgcn_wmma_f32_16x16x4_f32(false, a, false, bee, (short)0, aee, false, false);
            aeo = __builtin_amdgcn_wmma_f32_16x16x4_f32(false, a, false, beo, (short)0, aeo, false, false);
            aoe = __builtin_amdgcn_wmma_f32_16x16x4_f32(false, a, false, boe, (short)0, aoe, false, false);
            aoo = __builtin_amdgcn_wmma_f32_16x16x4_f32(false, a, false, boo, (short)0, aoo, false, false);
        }
        if (col < 9) {
            int n2 = n << 1;
#pragma unroll
            for (int q2 = 0; q2 < 8; ++q2) {
                int m = m0 + q2 + (hi ? 8 : 0);
                int t = m & 63;
                int r = m >> 6;
                int j = r % n; r /= n;
                int i = r % n;
                int b = r / n;
                size_t b00 = (((size_t)(b * n2 + 2 * i) * n2 + 2 * j) * 64 + t) * 9 + col;
                size_t b01 = (((size_t)(b * n2 + 2 * i) * n2 + 2 * j + 1) * 64 + t) * 9 + col;
                size_t b10 = (((size_t)(b * n2 + 2 * i + 1) * n2 + 2 * j) * 64 + t) * 9 + col;
                size_t b11 = (((size_t)(b * n2 + 2 * i + 1) * n2 + 2 * j + 1) * 64 + t) * 9 + col;
                out[b00] = aee[q2];
                out[b01] = aeo[q2];
                out[b10] = aoe[q2];
                out[b11] = aoo[q2];
            }
        }
    }
}

// ---------------------------------------------------------------------------
// sparseKernel3d: 3x3x3 conv (9->9, SAME) + bias + ReLU + Linear 9->9.
// Weights staged in LDS.  One thread per (b,x,y,t).
// ---------------------------------------------------------------------------
__global__ void k_conv9(const float* __restrict__ in,
                        const float* __restrict__ cw,
                        const float* __restrict__ cb,
                        const float* __restrict__ lw,
                        const float* __restrict__ lb,
                        float* __restrict__ out,
                        int n, int total, int accumulate) {
    __shared__ float scw[2187];
    __shared__ float slw[81];
    __shared__ float scb[9];
    __shared__ float slb[9];
    for (int idx = threadIdx.x; idx < 2187; idx += blockDim.x) scw[idx] = cw[idx];
    if (threadIdx.x < 81) slw[threadIdx.x] = lw[threadIdx.x];
    if (threadIdx.x < 9) { scb[threadIdx.x] = cb[threadIdx.x]; slb[threadIdx.x] = lb[threadIdx.x]; }
    __syncthreads();

    int gid = blockIdx.x * blockDim.x + threadIdx.x;
    if (gid >= total) return;
    int t = gid & 63;
    int r = gid >> 6;
    int y = r % n; r /= n;
    int x = r % n;
    int b = r / n;

    float acc[9];
#pragma unroll
    for (int o = 0; o < 9; ++o) acc[o] = 0.f;

    for (int dz = 0; dz < 3; ++dz) {
        int xx = x + dz - 1;
        if ((unsigned)xx >= (unsigned)n) continue;
        for (int dy = 0; dy < 3; ++dy) {
            int yy = y + dy - 1;
            if ((unsigned)yy >= (unsigned)n) continue;
            for (int dx = 0; dx < 3; ++dx) {
                int tt = t + dx - 1;
                if ((unsigned)tt >= 64u) continue;
                int tap = dz * 9 + dy * 3 + dx;
                const float* src = in + (((size_t)(b * n + xx) * n + yy) * 64 + tt) * 9;
#pragma unroll
                for (int i = 0; i < 9; ++i) {
                    float v = src[i];
#pragma unroll
                    for (int o = 0; o < 9; ++o)
                        acc[o] = fmaf(v, scw[(o * 9 + i) * 27 + tap], acc[o]);
                }
            }
        }
    }
    float rv[9];
#pragma unroll
    for (int o = 0; o < 9; ++o) rv[o] = fmaxf(acc[o] + scb[o], 0.f);
#pragma unroll
    for (int c = 0; c < 9; ++c) {
        float v = slb[c];
#pragma unroll
        for (int o = 0; o < 9; ++o) v = fmaf(rv[o], slw[c * 9 + o], v);
        size_t oi = (size_t)gid * 9 + c;
        if (accumulate) out[oi] += v; else out[oi] = v;
    }
}

// ---------------------------------------------------------------------------
// Truncated separable DFT chain for sparseKernelFT3d (only 4 corner blocks of
// <=5x5x5 modes are kept, so a direct mode-restricted DFT beats a full FFT).
// ---------------------------------------------------------------------------
__global__ void k_dft_t(const float* __restrict__ in, float* __restrict__ F1,
                        int n, int total) {
    int gid = blockIdx.x * blockDim.x + threadIdx.x;
    if (gid >= total) return;
    int kt = gid % 5; int r = gid / 5;
    int c = r % 9; r /= 9;
    int y = r % n; r /= n;
    int x = r % n; int b = r / n;
    float w = -TWO_PI * (float)kt / 64.f;
    float re = 0.f, im = 0.f;
    const float* base = in + (((size_t)(b * n + x) * n + y) * 64) * 9 + c;
    for (int t = 0; t < 64; ++t) {
        float v = base[(size_t)t * 9];
        float ang = w * (float)t;
        re = fmaf(v, __cosf(ang), re);
        im = fmaf(v, __sinf(ang), im);
    }
    size_t oi = ((((size_t)(b * n + x) * n + y) * 9 + c) * 5 + kt) * 2;
    F1[oi] = re; F1[oi + 1] = im;
}

__global__ void k_dft_y(const float* __restrict__ F1, float* __restrict__ F2,
                        int n, int L, int total) {
    int gid = blockIdx.x * blockDim.x + threadIdx.x;
    if (gid >= total) return;
    int kt = gid % 5; int r = gid / 5;
    int c = r % 9; r /= 9;
    int my = r % L; r /= L;
    int x = r % n; int b = r / n;
    int ky = mode_k(my, n);
    float w = -TWO_PI * (float)ky / (float)n;
    float re = 0.f, im = 0.f;
    for (int y = 0; y < n; ++y) {
        size_t ii = ((((size_t)(b * n + x) * n + y) * 9 + c) * 5 + kt) * 2;
        float ar = F1[ii], ai = F1[ii + 1];
        float ang = w * (float)y;
        float cs = __cosf(ang), sn = __sinf(ang);
        re += ar * cs - ai * sn;
        im += ar * sn + ai * cs;
    }
    size_t oi = ((((size_t)(b * n + x) * L + my) * 9 + c) * 5 + kt) * 2;
    F2[oi] = re; F2[oi + 1] = im;
}

__global__ void k_dft_x(const float* __restrict__ F2, float* __restrict__ XF,
                        int n, int L, int total) {
    int gid = blockIdx.x * blockDim.x + threadIdx.x;
    if (gid >= total) return;
    int kt = gid % 5; int r = gid / 5;
    int c = r % 9; r /= 9;
    int my = r % L; r /= L;
    int mx = r % L; int b = r / L;
    int kx = mode_k(mx, n);
    float w = -TWO_PI * (float)kx / (float)n;
    float re = 0.f, im = 0.f;
    for (int x = 0; x < n; ++x) {
        size_t ii = ((((size_t)(b * n + x) * L + my) * 9 + c) * 5 + kt) * 2;
        float ar = F2[ii], ai = F2[ii + 1];
        float ang = w * (float)x;
        float cs = __cosf(ang), sn = __sinf(ang);
        re += ar * cs - ai * sn;
        im += ar * sn + ai * cs;
    }
    size_t oi = ((((size_t)(b * L + mx) * L + my) * 9 + c) * 5 + kt) * 2;
    XF[oi] = re; XF[oi + 1] = im;
}

__global__ void k_modemix(const float* __restrict__ XF,
                          const float* __restrict__ w1, const float* __restrict__ w2,
                          const float* __restrict__ w3, const float* __restrict__ w4,
                          float* __restrict__ MX, int n, int L, int total) {
    int gid = blockIdx.x * blockDim.x + threadIdx.x;
    if (gid >= total) return;
    int kt = gid % 5; int r = gid / 5;
    int o = r % 9; r /= 9;
    int my = r % L; r /= L;
    int mx = r % L; int b = r / L;
    int kx = mode_k(mx, n), ky = mode_k(my, n);
    int l1 = (n / 2 + 1 < 5) ? (n / 2 + 1) : 5;
    bool xhi = kx >= n - l1, yhi = ky >= n - l1;
    const float* wsel; int lx, ly;
    if (xhi && yhi) { wsel = w4; lx = kx - (n - l1); ly = ky - (n - l1); }
    else if (yhi)   { wsel = w3; lx = kx;            ly = ky - (n - l1); }
    else if (xhi)   { wsel = w2; lx = kx - (n - l1); ly = ky;            }
    else            { wsel = w1; lx = kx;            ly = ky;            }
    float re = 0.f, im = 0.f;
#pragma unroll
    for (int i = 0; i < 9; ++i) {
        size_t ai_idx = ((((size_t)(b * L + mx) * L + my) * 9 + i) * 5 + kt) * 2;
        float ar = XF[ai_idx], ai = XF[ai_idx + 1];
        size_t wi_idx = ((((size_t)(i * 9 + o) * 5 + lx) * 5 + ly) * 5 + kt) * 2;
        float wr = wsel[wi_idx], wim = wsel[wi_idx + 1];
        re += ar * wr - ai * wim;
        im += ar * wim + ai * wr;
    }
    size_t oi = ((((size_t)(b * L + mx) * L + my) * 9 + o) * 5 + kt) * 2;
    MX[oi] = re; MX[oi + 1] = im;
}

__global__ void k_idft_x(const float* __restrict__ MX, float* __restrict__ I1,
                         int n, int L, int total) {
    int gid = blockIdx.x * blockDim.x + threadIdx.x;
    if (gid >= total) return;
    int kt = gid % 5; int r = gid / 5;
    int o = r % 9; r /= 9;
    int my = r % L; r /= L;
    int x = r % n; int b = r / n;
    float re = 0.f, im = 0.f;
    for (int mx = 0; mx < L; ++mx) {
        int kx = mode_k(mx, n);
        float ang = TWO_PI * (float)kx * (float)x / (float)n;
        float cs = __cosf(ang), sn = __sinf(ang);
        size_t ii = ((((size_t)(b * L + mx) * L + my) * 9 + o) * 5 + kt) * 2;
        float ar = MX[ii], ai = MX[ii + 1];
        re += ar * cs - ai * sn;
        im += ar * sn + ai * cs;
    }
    size_t oi = ((((size_t)(b * n + x) * L + my) * 9 + o) * 5 + kt) * 2;
    I1[oi] = re; I1[oi + 1] = im;
}

__global__ void k_idft_y(const float* __restrict__ I1, float* __restrict__ I2,
                         int n, int L, int total) {
    int gid = blockIdx.x * blockDim.x + threadIdx.x;
    if (gid >= total) return;
    int kt = gid % 5; int r = gid / 5;
    int o = r % 9; r /= 9;
    int y = r % n; r /= n;
    int x = r % n; int b = r / n;
    float re = 0.f, im = 0.f;
    for (int my = 0; my < L; ++my) {
        int ky = mode_k(my, n);
        float ang = TWO_PI * (float)ky * (float)y / (float)n;
        float cs = __cosf(ang), sn = __sinf(ang);
        size_t ii = ((((size_t)(b * n + x) * L + my) * 9 + o) * 5 + kt) * 2;
        float ar = I1[ii], ai = I1[ii + 1];
        re += ar * cs - ai * sn;
        im += ar * sn + ai * cs;
    }
    size_t oi = ((((size_t)(b * n + x) * n + y) * 9 + o) * 5 + kt) * 2;
    I2[oi] = re; I2[oi + 1] = im;
}

// irfft along t (only 5 modes, kt=0 DC) + 1/N + ReLU + Linear(A_Lo) -> Ud
__global__ void k_idft_t_lin(const float* __restrict__ I2,
                             const float* __restrict__ low,
                             const float* __restrict__ lob,
                             float* __restrict__ out, int n, int total) {
    int gid = blockIdx.x * blockDim.x + threadIdx.x;
    if (gid >= total) return;
    int t = gid & 63;
    int r = gid >> 6;
    int y = r % n; r /= n;
    int x = r % n; int b = r / n;
    float invN = 1.f / (float)(n * n * 64);
    float rv[9];
#pragma unroll
    for (int o = 0; o < 9; ++o) {
        size_t base = ((((size_t)(b * n + x) * n + y) * 9 + o) * 5) * 2;
        float v = I2[base];                         // Re(X_0)
#pragma unroll
        for (int k = 1; k < 5; ++k) {
            float ang = TWO_PI * (float)k * (float)t / 64.f;
            v += 2.f * (I2[base + 2 * k] * __cosf(ang) - I2[base + 2 * k + 1] * __sinf(ang));
        }
        rv[o] = fmaxf(v * invN, 0.f);
    }
#pragma unroll
    for (int c = 0; c < 9; ++c) {
        float v = lob[c];
#pragma unroll
        for (int o = 0; o < 9; ++o) v = fmaf(rv[o], low[c * 9 + o], v);
        out[(size_t)gid * 9 + c] = v;
    }
}

// Coarsest-level Linear T0 (9->9) at n0=1
__global__ void k_t0(const float* __restrict__ xin, const float* __restrict__ w,
                     const float* __restrict__ bb, float* __restrict__ o, int total) {
    int gid = blockIdx.x * blockDim.x + threadIdx.x;
    if (gid >= total) return;
    const float* src = xin + (size_t)gid * 9;
    float v[9];
#pragma unroll
    for (int i = 0; i < 9; ++i) v[i] = src[i];
#pragma unroll
    for (int c = 0; c < 9; ++c) {
        float s = bb[c];
#pragma unroll
        for (int k = 0; k < 9; ++k) s = fmaf(v[k], w[c * 9 + k], s);
        o[(size_t)gid * 9 + c] = s;
    }
}

static inline int NB(int t) { return (t + 255) / 256; }

extern "C" void kernel_launch(void* const* d_in, const int* in_sizes, int n_in,
                              void* d_out, int out_size, void* d_ws, size_t ws_size,
                              hipStream_t stream) {
    (void)in_sizes; (void)n_in; (void)out_size; (void)ws_size;
    const float* x     = (const float*)d_in[0];
    const float* ec_s  = (const float*)d_in[1];
    const float* ec_d  = (const float*)d_in[2];
    const float* rc_ee = (const float*)d_in[3];
    const float* rc_eo = (const float*)d_in[4];
    const float* rc_oe = (const float*)d_in[5];
    const float* rc_oo = (const float*)d_in[6];
    const float* w1    = (const float*)d_in[7];
    const float* w2    = (const float*)d_in[8];
    const float* w3    = (const float*)d_in[9];
    const float* w4    = (const float*)d_in[10];
    const float* ALo_w = (const float*)d_in[11];
    const float* ALo_b = (const float*)d_in[12];
    const float* Bcw   = (const float*)d_in[13];
    const float* Bcb   = (const float*)d_in[14];
    const float* Blw   = (const float*)d_in[15];
    const float* Blb   = (const float*)d_in[16];
    const float* Ccw   = (const float*)d_in[17];
    const float* Ccb   = (const float*)d_in[18];
    const float* Clw   = (const float*)d_in[19];
    const float* Clb   = (const float*)d_in[20];
    const float* T0w   = (const float*)d_in[21];
    const float* T0b   = (const float*)d_in[22];
    float* out = (float*)d_out;
    float* ws  = (float*)d_ws;

    static const int LVLN[6] = {32, 16, 8, 4, 2, 1};
    // element counts per level: 8*64*9*n*n
    size_t lvl_off[6]; size_t acc = 0;
    for (int k = 0; k < 6; ++k) { lvl_off[k] = acc; acc += (size_t)4608 * LVLN[k] * LVLN[k]; }
    const size_t PYR = acc;                       // 6,289,920 floats

    size_t Ud_off = 0;
    size_t Us_off = Ud_off + PYR;
    size_t d_off  = Us_off + PYR;                 // max-level d scratch (4,718,592)
    size_t xA_off = d_off + 4718592;
    size_t xB_off = xA_off + 4718592;
    size_t fa_off = xB_off + 1179648;
    size_t fb_off = fa_off + 737280;              // total ~24.7M floats (~99 MB)

    float* Ud = ws + Ud_off;
    float* Us = ws + Us_off;
    float* dbuf = ws + d_off;
    float* xsbuf[2] = { ws + xA_off, ws + xB_off };
    float* fa = ws + fa_off;
    float* fb = ws + fb_off;

    dim3 blk(256);
    const float* xin = x;
    for (int k = 0; k < 6; ++k) {
        int n = LVLN[k];
        int Mtot = 8 * n * n * 64;
        int tiles = Mtot / 16;
        float* sbuf = xsbuf[k & 1];
        float* Udl = Ud + lvl_off[k];
        float* Usl = Us + lvl_off[k];
        int L = (n <= 10) ? n : 10;

        k_wavelet<<<(tiles + 7) / 8, blk, 0, stream>>>(xin, ec_d, ec_s, dbuf, sbuf, n, tiles);

        // spectral kernel on d -> Udl
        int t1 = 8 * n * n * 9 * 5;
        k_dft_t<<<NB(t1), blk, 0, stream>>>(dbuf, fa, n, t1);
        int t2 = 8 * n * L * 9 * 5;
        k_dft_y<<<NB(t2), blk, 0, stream>>>(fa, fb, n, L, t2);
        int t3 = 8 * L * L * 9 * 5;
        k_dft_x<<<NB(t3), blk, 0, stream>>>(fb, fa, n, L, t3);
        k_modemix<<<NB(t3), blk, 0, stream>>>(fa, w1, w2, w3, w4, fb, n, L, t3);
        k_idft_x<<<NB(t2), blk, 0, stream>>>(fb, fa, n, L, t2);
        k_idft_y<<<NB(t1), blk, 0, stream>>>(fa, fb, n, L, t1);
        k_idft_t_lin<<<NB(Mtot), blk, 0, stream>>>(fb, ALo_w, ALo_b, Udl, n, Mtot);

        // Ud += conv_B(s);  Us = conv_C(d)
        k_conv9<<<NB(Mtot), blk, 0, stream>>>(sbuf, Bcw, Bcb, Blw, Blb, Udl, n, Mtot, 1);
        k_conv9<<<NB(Mtot), blk, 0, stream>>>(dbuf, Ccw, Ccb, Clw, Clb, Usl, n, Mtot, 0);

        xin = sbuf;
    }

    // coarsest linear T0 (n0 = 1): 8*64 rows
    float* rA = ws + d_off;       // reuse d scratch for reconstruction ping
    float* rB = ws + xA_off;      // reuse xA for reconstruction pong
    k_t0<<<NB(512), blk, 0, stream>>>(xin, T0w, T0b, rA, 512);

    const float* cur = rA;
    bool curIsA = true;
    for (int i = 5; i >= 0; --i) {
        int n = LVLN[i];
        int Mtot = 8 * n * n * 64;
        int tiles = Mtot / 16;
        float* dst = (i == 0) ? out : (curIsA ? rB : rA);
        k_evenodd<<<(tiles + 7) / 8, blk, 0, stream>>>(cur, Us + lvl_off[i], Ud + lvl_off[i],
                                                       rc_ee, rc_eo, rc_oe, rc_oo, dst, n, tiles);
        cur = dst;
        curIsA = !curIsA;
    }
}